// PatchEmbed_27255862461160
// MI455X (gfx1250) — compile-verified
//
#include <hip/hip_runtime.h>
#include <cstdint>

// PatchEmbed: MaxPool3d(k=4,s=2) on [128,128,128,16] f32 -> pooled [63,63,63,16],
// then overlapping 4^3 patch extraction (30^3 blocks, stride 2) ->
// out [27000, 64, 16] f32.  Pure data movement + max reduction: memory-bound,
// ~245MB min HBM traffic -> ~11us floor at 23.3 TB/s.  No matmul => no WMMA;
// CDNA5 feature used: async global->LDS copies (ASYNCcnt) for tile staging,
// plus non-temporal streaming stores for the write-once output.

typedef float v4f __attribute__((ext_vector_type(4)));
typedef int   v4i __attribute__((ext_vector_type(4)));
typedef __attribute__((address_space(1))) v4i* gptr_v4i;  // global int4*
typedef __attribute__((address_space(3))) v4i* lptr_v4i;  // LDS int4*

#define POOL_D 63   // pooled spatial extent per dim: (128-4)/2+1
#define NB     30   // patch blocks per dim: (63-4)/2+1
#define NWG    15   // workgroups per dim in gather pass (2x2x2 blocks each)

#if defined(__has_builtin)
#if __has_builtin(__builtin_amdgcn_global_load_async_to_lds_b128)
#define HAVE_ASYNC_LDS 1
#endif
#if __has_builtin(__builtin_amdgcn_s_wait_asynccnt)
#define HAVE_WAIT_ASYNC 1
#endif
#endif

__device__ __forceinline__ v4f vmax4(v4f a, v4f b) {
  v4f r;
  r.x = fmaxf(a.x, b.x);
  r.y = fmaxf(a.y, b.y);
  r.z = fmaxf(a.z, b.z);
  r.w = fmaxf(a.w, b.w);
  return r;
}

// ---------------------------------------------------------------------------
// Pass 1: 3D max pool.  One thread per (pooled voxel, 4-channel group).
// x layout [d][h][w][c], c innermost; v4f strides: d=65536, h=512, w=4, c4=1.
// x is read with 8x window overlap, but the whole 134MB volume fits in the
// 192MB L2, so HBM sees it ~once.
// ---------------------------------------------------------------------------
__global__ __launch_bounds__(256) void pool3d_kernel(const float* __restrict__ x,
                                                     float* __restrict__ pooled) {
  const int NV = POOL_D * POOL_D * POOL_D;  // 250047 voxels
  int t = blockIdx.x * 256 + threadIdx.x;
  if (t >= NV * 4) return;
  int c4 = t & 3;
  int v  = t >> 2;
  int wp = v % POOL_D;
  int q  = v / POOL_D;
  int hp = q % POOL_D;
  int dp = q / POOL_D;

  const v4f* xf = (const v4f*)x;
  int base = (dp * 2) * 65536 + (hp * 2) * 512 + (wp * 2) * 4 + c4;

  v4f m = xf[base];
#pragma unroll
  for (int dd = 0; dd < 4; ++dd) {
#pragma unroll
    for (int hh = 0; hh < 4; ++hh) {
#pragma unroll
      for (int ww = 0; ww < 4; ++ww) {
        if (dd == 0 && hh == 0 && ww == 0) continue;
        m = vmax4(m, xf[base + dd * 65536 + hh * 512 + ww * 4]);
      }
    }
  }
  ((v4f*)pooled)[v * 4 + c4] = m;  // pooled layout [dp][hp][wp][c]
}

// ---------------------------------------------------------------------------
// Pass 2: patch gather.  Workgroup = 2x2x2 tile of output blocks; its source
// is a 6x6x6x16 pooled region (13.8 KB) staged into LDS via async copies
// (each pooled voxel feeds up to 8 blocks), then 8 blocks x 4KB of coalesced
// non-temporal float4 stores.
// out[((I*30+J)*30+K), pi*16+pj*4+pk, c] = pooled[2I+pi, 2J+pj, 2K+pk, c]
// ---------------------------------------------------------------------------
__global__ __launch_bounds__(256) void gather_kernel(const float* __restrict__ pooled,
                                                     float* __restrict__ out) {
  __shared__ float lds[6 * 6 * 6 * 16];  // 13824 bytes

  int wg = blockIdx.x;                    // 0 .. 15^3-1
  int gk = wg % NWG;
  int q  = wg / NWG;
  int gj = q % NWG;
  int gi = q / NWG;
  int d0 = 4 * gi, h0 = 4 * gj, w0 = 4 * gk;  // pooled-region origin

  // --- stage 6*6*6 voxels * 4 float4 = 864 b128 copies into LDS ---
  for (int idx = threadIdx.x; idx < 864; idx += 256) {
    int c4 = idx & 3;
    int vv = idx >> 2;          // 0..215
    int ww = vv % 6;
    int q2 = vv / 6;
    int hh = q2 % 6;
    int dd = q2 / 6;
    int gfi = (((d0 + dd) * POOL_D + (h0 + hh)) * POOL_D + (w0 + ww)) * 16 + c4 * 4;
    float* l = &lds[vv * 16 + c4 * 4];
#if HAVE_ASYNC_LDS
    __builtin_amdgcn_global_load_async_to_lds_b128(
        (gptr_v4i)(uintptr_t)(pooled + gfi),
        (lptr_v4i)(uintptr_t)l,
        0, 0);
#else
    *(v4f*)l = *(const v4f*)(pooled + gfi);
#endif
  }

#if HAVE_ASYNC_LDS
#if HAVE_WAIT_ASYNC
  __builtin_amdgcn_s_wait_asynccnt(0);
#else
  asm volatile("s_wait_asynccnt 0" ::: "memory");
#endif
#endif
  __syncthreads();

  // --- write 8 blocks * 64 positions * 4 float4 = 2048 stores ---
  const v4f* lf = (const v4f*)lds;
  v4f* of = (v4f*)out;
  for (int idx = threadIdx.x; idx < 2048; idx += 256) {
    int c4  = idx & 3;
    int pos = (idx >> 2) & 63;  // pi*16 + pj*4 + pk
    int blk = idx >> 8;         // bi*4 + bj*2 + bk
    int pk = pos & 3, pj = (pos >> 2) & 3, pi = pos >> 4;
    int bk = blk & 1, bj = (blk >> 1) & 1, bi = blk >> 2;
    int I = 2 * gi + bi, J = 2 * gj + bj, K = 2 * gk + bk;
    int n = (I * NB + J) * NB + K;
    int dd = 2 * bi + pi, hh = 2 * bj + pj, ww = 2 * bk + pk;
    v4f val = lf[((dd * 6 + hh) * 6 + ww) * 4 + c4];
    __builtin_nontemporal_store(val, &of[(n * 64 + pos) * 4 + c4]);
  }
}

extern "C" void kernel_launch(void* const* d_in, const int* in_sizes, int n_in,
                              void* d_out, int out_size, void* d_ws, size_t ws_size,
                              hipStream_t stream) {
  (void)in_sizes; (void)n_in; (void)out_size; (void)ws_size;
  const float* x = (const float*)d_in[0];
  float* pooled  = (float*)d_ws;   // needs 63^3*16*4 = 16,003,008 bytes
  float* out     = (float*)d_out;  // 27000*64*16 floats

  int total1 = POOL_D * POOL_D * POOL_D * 4;  // 1,000,188 threads
  pool3d_kernel<<<(total1 + 255) / 256, 256, 0, stream>>>(x, pooled);
  gather_kernel<<<NWG * NWG * NWG, 256, 0, stream>>>(pooled, out);
}